// AtomMPNN_90683939487977
// MI455X (gfx1250) — compile-verified
//
#include <hip/hip_runtime.h>
#include <math.h>

typedef __attribute__((ext_vector_type(2))) float v2f;
typedef __attribute__((ext_vector_type(8))) float v8f;

#define NB 8
#define NA 8192
#define NK 32
#define ND 64
#define NIN 129
#define HSTRIDE 68

__device__ __forceinline__ v8f wmma_f32_16x16x4(v2f a, v2f b, v8f c) {
  return __builtin_amdgcn_wmma_f32_16x16x4_f32(false, a, false, b, (short)0, c,
                                               false, false);
}

__device__ __forceinline__ float gelu_exact(float x) {
  return 0.5f * x * (1.0f + erff(x * 0.70710678118654752440f));
}

__global__ void __launch_bounds__(256) mpnn_msg_kernel(
    const float* __restrict__ emb, const float* __restrict__ dists,
    const int* __restrict__ eidx, const float* __restrict__ mask,
    const float* __restrict__ W0, const float* __restrict__ b0,
    const float* __restrict__ W1, const float* __restrict__ b1,
    const float* __restrict__ W2, const float* __restrict__ b2,
    float* __restrict__ upd) {
  __shared__ float ybias[16 * ND];            // per-atom self-term + b0
  __shared__ float hscr[8][16 * HSTRIDE];     // per-wave transpose scratch

  const int blk = blockIdx.x;
  const int b = blk >> 9;                 // 512 blocks per batch
  const int abase = (blk & 511) << 4;     // 16 atoms per block
  const int tid = threadIdx.x;
  const int lane = tid & 31;
  const int wave = tid >> 5;
  const int rl = lane & 15;
  const int hl = lane >> 4;
  const long long bN = (long long)b * NA;

  // ---- stage 0: y_self[a][n] = sum_k W0[n][64+k]*emb_m[a][k] + b0[n], wave 0 only
  if (wave == 0) {
    const int arow = abase + rl;
    const float ma = mask[bN + arow];
    const float* erow = emb + (bN + arow) * (long long)ND;
    v8f acc[4];
#pragma unroll
    for (int nt = 0; nt < 4; ++nt)
#pragma unroll
      for (int v = 0; v < 8; ++v) acc[nt][v] = 0.0f;
#pragma unroll
    for (int kk = 0; kk < 16; ++kk) {
      const float2 av = *reinterpret_cast<const float2*>(erow + 4 * kk + 2 * hl);
      v2f af; af.x = av.x * ma; af.y = av.y * ma;
#pragma unroll
      for (int nt = 0; nt < 4; ++nt) {
        const int col = nt * 16 + rl;
        const float* wp = W0 + col * NIN + 64 + 4 * kk + 2 * hl;
        v2f bf; bf.x = wp[0]; bf.y = wp[1];
        acc[nt] = wmma_f32_16x16x4(af, bf, acc[nt]);
      }
    }
#pragma unroll
    for (int nt = 0; nt < 4; ++nt) {
      const int col = nt * 16 + rl;
      const float bb = b0[col];
#pragma unroll
      for (int v = 0; v < 8; ++v)
        ybias[(v + 8 * hl) * ND + col] = acc[nt][v] + bb;
    }
  }
  __syncthreads();

  // ---- stage 1: per-wave, 2 atoms, 2 half-tiles (16 edges) each
  float msg[4] = {0.f, 0.f, 0.f, 0.f};
  int nvacc = 0;
  float* hs = hscr[wave];

  for (int t = 0; t < 4; ++t) {
    const int aloc = 2 * wave + (t >> 1);
    const int half = t & 1;
    const int a = abase + aloc;
    const long long ebase = (bN + a) * NK + half * 16;

    const int ei = eidx[ebase + rl];
    const int validi = (ei >= 0) ? 1 : 0;
    const int nbr = validi ? ei : 0;
    const float mn = validi ? mask[bN + nbr] : 0.0f;
    const float dl = dists[ebase + rl];
    const float vfl = validi ? 1.0f : 0.0f;
    float dr[8], vr[8];
#pragma unroll
    for (int v = 0; v < 8; ++v) {
      dr[v] = __shfl(dl, v + 8 * hl, 32);
      vr[v] = __shfl(vfl, v + 8 * hl, 32);
    }

    v8f acc[4];
#pragma unroll
    for (int nt = 0; nt < 4; ++nt)
#pragma unroll
      for (int v = 0; v < 8; ++v) acc[nt][v] = 0.0f;

    // ---- layer 0: K=64 over gathered neighbor features
    const float* srow = emb + (bN + nbr) * (long long)ND;
#pragma unroll
    for (int kk = 0; kk < 16; ++kk) {
      const float2 av = *reinterpret_cast<const float2*>(srow + 4 * kk + 2 * hl);
      v2f af; af.x = av.x * mn; af.y = av.y * mn;
#pragma unroll
      for (int nt = 0; nt < 4; ++nt) {
        const int col = nt * 16 + rl;
        const float* wp = W0 + col * NIN + 4 * kk + 2 * hl;
        v2f bf; bf.x = wp[0]; bf.y = wp[1];
        acc[nt] = wmma_f32_16x16x4(af, bf, acc[nt]);
      }
    }
    // bias (self-term + dist term) + GELU, transpose to LDS
#pragma unroll
    for (int nt = 0; nt < 4; ++nt) {
      const int col = nt * 16 + rl;
      const float yb = ybias[aloc * ND + col];
      const float w0d = W0[col * NIN + 128];
#pragma unroll
      for (int v = 0; v < 8; ++v) {
        const float x = acc[nt][v] + yb + w0d * dr[v];
        hs[(v + 8 * hl) * HSTRIDE + col] = gelu_exact(x);
      }
    }
    __builtin_amdgcn_wave_barrier();
    asm volatile("s_wait_dscnt 0" ::: "memory");

    // ---- layer 1
#pragma unroll
    for (int nt = 0; nt < 4; ++nt) {
      const float bb = b1[nt * 16 + rl];
#pragma unroll
      for (int v = 0; v < 8; ++v) acc[nt][v] = bb;
    }
#pragma unroll
    for (int kk = 0; kk < 16; ++kk) {
      v2f af;
      af.x = hs[rl * HSTRIDE + 4 * kk + 2 * hl];
      af.y = hs[rl * HSTRIDE + 4 * kk + 2 * hl + 1];
#pragma unroll
      for (int nt = 0; nt < 4; ++nt) {
        const int col = nt * 16 + rl;
        const float2 wv =
            *reinterpret_cast<const float2*>(W1 + col * ND + 4 * kk + 2 * hl);
        v2f bf; bf.x = wv.x; bf.y = wv.y;
        acc[nt] = wmma_f32_16x16x4(af, bf, acc[nt]);
      }
    }
    __builtin_amdgcn_wave_barrier();
#pragma unroll
    for (int nt = 0; nt < 4; ++nt) {
      const int col = nt * 16 + rl;
#pragma unroll
      for (int v = 0; v < 8; ++v)
        hs[(v + 8 * hl) * HSTRIDE + col] = gelu_exact(acc[nt][v]);
    }
    __builtin_amdgcn_wave_barrier();
    asm volatile("s_wait_dscnt 0" ::: "memory");

    // ---- layer 2
#pragma unroll
    for (int nt = 0; nt < 4; ++nt) {
      const float bb = b2[nt * 16 + rl];
#pragma unroll
      for (int v = 0; v < 8; ++v) acc[nt][v] = bb;
    }
#pragma unroll
    for (int kk = 0; kk < 16; ++kk) {
      v2f af;
      af.x = hs[rl * HSTRIDE + 4 * kk + 2 * hl];
      af.y = hs[rl * HSTRIDE + 4 * kk + 2 * hl + 1];
#pragma unroll
      for (int nt = 0; nt < 4; ++nt) {
        const int col = nt * 16 + rl;
        const float2 wv =
            *reinterpret_cast<const float2*>(W2 + col * ND + 4 * kk + 2 * hl);
        v2f bf; bf.x = wv.x; bf.y = wv.y;
        acc[nt] = wmma_f32_16x16x4(af, bf, acc[nt]);
      }
    }
    __builtin_amdgcn_wave_barrier();

    // GELU, zero invalid edges, column-sum (sum over 16 edge rows)
    float cs[4];
#pragma unroll
    for (int nt = 0; nt < 4; ++nt) {
      float s = 0.0f;
#pragma unroll
      for (int v = 0; v < 8; ++v) s += gelu_exact(acc[nt][v]) * vr[v];
      s += __shfl_xor(s, 16, 32);  // combine row halves across lane halves
      cs[nt] = s;
    }
    const unsigned long long bal = __ballot(validi);
    const int nvt = __popcll(bal & 0xFFFFull);

    if (half == 0) {
      msg[0] = cs[0]; msg[1] = cs[1]; msg[2] = cs[2]; msg[3] = cs[3];
      nvacc = nvt;
    } else {
      float nvf = (float)(nvacc + nvt);
      if (nvf == 0.0f) nvf = 1.0f;
      const float ma = mask[bN + a];
      const long long obase = (bN + a) * (long long)ND;
      const int c0 = rl + 32 * hl;
      const float m0 = hl ? (msg[2] + cs[2]) : (msg[0] + cs[0]);
      const float m1 = hl ? (msg[3] + cs[3]) : (msg[1] + cs[1]);
      upd[obase + c0] = (emb[obase + c0] + m0 / nvf) * ma;
      upd[obase + c0 + 16] = (emb[obase + c0 + 16] + m1 / nvf) * ma;
    }
  }
}

// ---- per-(b,d) sum / sumsq / count -> mean, inv_std
__global__ void __launch_bounds__(256) mpnn_stats_kernel(
    const float* __restrict__ upd, const float* __restrict__ mask,
    float* __restrict__ stats) {
  const int b = blockIdx.x;
  const int tid = threadIdx.x;
  const int ch = tid & 63;
  const int g = tid >> 6;
  const size_t base = (size_t)b * NA * ND;
  float S = 0.0f, SS = 0.0f, C = 0.0f;
  for (int a = g; a < NA; a += 4) {
    const float v = upd[base + (size_t)a * ND + ch];
    S += v;
    SS += v * v;
    if (ch == 0) C += mask[(size_t)b * NA + a];
  }
  __shared__ float sS[256], sQ[256], sC[4];
  sS[tid] = S; sQ[tid] = SS;
  if (ch == 0) sC[g] = C;
  __syncthreads();
  if (tid < 64) {
    const float St = sS[tid] + sS[tid + 64] + sS[tid + 128] + sS[tid + 192];
    const float Qt = sQ[tid] + sQ[tid + 64] + sQ[tid + 128] + sQ[tid + 192];
    float cnt = sC[0] + sC[1] + sC[2] + sC[3];
    if (cnt == 0.0f) cnt = 1.0f;
    const float mean = St / cnt;
    // sum over ALL N atoms of (mf-mean)^2 = SS - 2*mean*S + N*mean^2
    const float var = (Qt - 2.0f * mean * St + (float)NA * mean * mean) / cnt;
    stats[b * ND + tid] = mean;
    stats[NB * ND + b * ND + tid] = 1.0f / sqrtf(var + 1e-5f);
  }
}

// ---- elementwise normalization, in place on d_out
__global__ void __launch_bounds__(256) mpnn_norm_kernel(
    float* __restrict__ out, const float* __restrict__ mask,
    const float* __restrict__ stats, const float* __restrict__ scale,
    const float* __restrict__ shift) {
  const size_t i = (size_t)blockIdx.x * 256 + threadIdx.x;
  const int ch = (int)(i & 63);
  const size_t an = i >> 6;       // b*N + a
  const int b = (int)(an >> 13);  // /8192
  const float mean = stats[b * ND + ch];
  const float inv = stats[NB * ND + b * ND + ch];
  float v = out[i];
  v = (v - mean) * inv * scale[ch] + shift[ch];
  out[i] = v * mask[an];
}

extern "C" void kernel_launch(void* const* d_in, const int* in_sizes, int n_in,
                              void* d_out, int out_size, void* d_ws,
                              size_t ws_size, hipStream_t stream) {
  const float* emb   = (const float*)d_in[0];
  const float* dists = (const float*)d_in[1];
  const int*   eix   = (const int*)d_in[2];
  const float* mask  = (const float*)d_in[3];
  const float* W0    = (const float*)d_in[4];
  const float* b0    = (const float*)d_in[5];
  const float* W1    = (const float*)d_in[6];
  const float* b1    = (const float*)d_in[7];
  const float* W2    = (const float*)d_in[8];
  const float* b2    = (const float*)d_in[9];
  const float* scale = (const float*)d_in[10];
  const float* shift = (const float*)d_in[11];
  float* out = (float*)d_out;
  float* stats = (float*)d_ws;  // 2 * NB * ND floats

  mpnn_msg_kernel<<<NB * (NA / 16), 256, 0, stream>>>(
      emb, dists, eix, mask, W0, b0, W1, b1, W2, b2, out);
  mpnn_stats_kernel<<<NB, 256, 0, stream>>>(out, mask, stats);
  const int total = NB * NA * ND;
  mpnn_norm_kernel<<<total / 256, 256, 0, stream>>>(out, mask, stats, scale,
                                                    shift);
}